// CategoricalDiffusionGrayscale_1606317768911
// MI455X (gfx1250) — compile-verified
//
#include <hip/hip_runtime.h>
#include <stdint.h>

// ---------------------------------------------------------------------------
// Categorical diffusion forward (gumbel-softmax sample), MI455X / gfx1250.
//
// Shapes (fixed by reference): x0 [32,64,64,256] f32 one-hot, beta [1000] f32,
// t scalar int. Output [32,64,64,256] f32.
//
// Math: bar_Q = abar*I + (1-abar)*J/256 and one-hot rows sum to 1, so
//   p = abar*x0 + (1-abar)/256           (one FMA per element)
//   out = softmax(log p + gumbel) = (p/e) / sum_row(p/e),  e = -log(u)
// u reproduces jax.random.uniform(key(1), ...) via Threefry-2x32, with rows
// r and r+65536 sharing one cipher call (JAX splits the iota in half).
// ---------------------------------------------------------------------------

typedef float v2f __attribute__((ext_vector_type(2)));
typedef float v8f __attribute__((ext_vector_type(8)));

#define K_CAT       256
#define ROWS_TOTAL  131072      // 32*64*64
#define HALF_ROWS   65536       // pairing offset: flat + 2^24 == row + 65536
#define LDS_STR     260         // padded LDS row stride (floats) -> no bank conflicts
#define WAVES_PER_BLOCK 2
#define PAIRS_PER_WAVE  8       // 8 low rows + 8 partner rows = 16 LDS rows / wave

__device__ __forceinline__ uint32_t rotl32(uint32_t x, int r) {
  return (x << r) | (x >> (32 - r));
}

// JAX threefry2x32 with key(1): k0=0, k1=1, k2 = 0^1^0x1BD11BDA = 0x1BD11BDB
__device__ __forceinline__ void threefry2x32(uint32_t X0, uint32_t X1,
                                             uint32_t& O0, uint32_t& O1) {
  const uint32_t K0 = 0u, K1 = 1u, K2 = 0x1BD11BDBu;
  uint32_t x0 = X0 + K0;
  uint32_t x1 = X1 + K1;
#define TF_R(r) { x0 += x1; x1 = rotl32(x1, r); x1 ^= x0; }
  TF_R(13) TF_R(15) TF_R(26) TF_R(6)   x0 += K1; x1 += K2 + 1u;
  TF_R(17) TF_R(29) TF_R(16) TF_R(24)  x0 += K2; x1 += K0 + 2u;
  TF_R(13) TF_R(15) TF_R(26) TF_R(6)   x0 += K0; x1 += K1 + 3u;
  TF_R(17) TF_R(29) TF_R(16) TF_R(24)  x0 += K1; x1 += K2 + 4u;
  TF_R(13) TF_R(15) TF_R(26) TF_R(6)   x0 += K2; x1 += K0 + 5u;
#undef TF_R
  O0 = x0; O1 = x1;
}

// jax.random.uniform(..., minval=1e-9, maxval=1.0) bit semantics.
__device__ __forceinline__ float bits_to_unit(uint32_t b) {
  float f = __uint_as_float((b >> 9) | 0x3F800000u) - 1.0f;  // [0,1)
  return fmaxf(1e-9f, f + 1e-9f);   // (1.0-1e-9) rounds to 1.0f in fp32
}

// One paired element: flat index of the low-half element; its partner is
// flat + 2^24 (same column, row + 65536). Produces both softmax numerators.
__device__ __forceinline__ void pair_elem(uint32_t flat, float vA, float vB,
                                          float abar, float pc,
                                          float& qA, float& qB) {
  uint32_t b0, b1;
  threefry2x32(flat, flat + 0x1000000u, b0, b1);
  float u0 = bits_to_unit(b0);
  float u1 = bits_to_unit(b1);
  float e0 = -__logf(u0);                 // exponential variate, > 0
  float e1 = -__logf(u1);
  float p0 = fmaf(abar, vA, pc);          // abar*x0 + (1-abar)/256
  float p1 = fmaf(abar, vB, pc);
  qA = p0 * __builtin_amdgcn_rcpf(e0);    // p * exp(gumbel)
  qB = p1 * __builtin_amdgcn_rcpf(e1);
}

// ---------------------------------------------------------------------------
// Kernel 1: abar = prod_{i<t} (1 - beta[i])
// ---------------------------------------------------------------------------
__global__ void abar_kernel(const float* __restrict__ beta,
                            const int* __restrict__ tptr,
                            int nbeta, float* __restrict__ ws) {
  __shared__ float red[256];
  int t = *tptr;
  if (t > nbeta) t = nbeta;
  if (t < 0) t = 0;
  float m = 1.0f;
  for (int j = (int)threadIdx.x; j < t; j += 256) m *= (1.0f - beta[j]);
  red[threadIdx.x] = m;
  __syncthreads();
  for (int s = 128; s > 0; s >>= 1) {
    if ((int)threadIdx.x < s) red[threadIdx.x] *= red[threadIdx.x + s];
    __syncthreads();
  }
  if (threadIdx.x == 0) ws[0] = red[0];
}

// ---------------------------------------------------------------------------
// Kernel 2: fused affine + gumbel + softmax. One wave owns 16 rows
// (8 low rows r0..r0+7 and their threefry partners r0+65536..+7).
// ---------------------------------------------------------------------------
__global__ __launch_bounds__(64) void catdiff_softmax_kernel(
    const float* __restrict__ x0, const float* __restrict__ abar_ws,
    float* __restrict__ out) {
  extern __shared__ float lds[];
  const int lane = (int)(threadIdx.x & 31u);
  const int wave = (int)(threadIdx.x >> 5u);
  float* q = lds + wave * (16 * LDS_STR);

  const int waveIdx = (int)blockIdx.x * WAVES_PER_BLOCK + wave;  // 0..8191
  const int r0 = waveIdx * PAIRS_PER_WAVE;                       // base low row
  const float abar = abar_ws[0];
  const float pc = (1.0f - abar) * (1.0f / 256.0f);

  const int c0 = lane * 4;        // column group 0 (float4)
  const int c1 = c0 + 128;        // column group 1 (float4)

  // ---- Pass 1: numerators q = p * exp(gumbel) into LDS -------------------
  for (int pr = 0; pr < PAIRS_PER_WAVE; ++pr) {
    const uint32_t baseA = (uint32_t)(r0 + pr) * K_CAT;            // < 2^24
    const uint32_t baseB = baseA + (uint32_t)HALF_ROWS * K_CAT;    // + 2^24
    const float4 a0 = *(const float4*)(x0 + baseA + c0);
    const float4 a1 = *(const float4*)(x0 + baseA + c1);
    const float4 b0 = *(const float4*)(x0 + baseB + c0);
    const float4 b1 = *(const float4*)(x0 + baseB + c1);
    const float av0[4] = {a0.x, a0.y, a0.z, a0.w};
    const float av1[4] = {a1.x, a1.y, a1.z, a1.w};
    const float bv0[4] = {b0.x, b0.y, b0.z, b0.w};
    const float bv1[4] = {b1.x, b1.y, b1.z, b1.w};
    float qa0[4], qa1[4], qb0[4], qb1[4];
#pragma unroll
    for (int e = 0; e < 4; ++e) {
      pair_elem(baseA + (uint32_t)(c0 + e), av0[e], bv0[e], abar, pc, qa0[e], qb0[e]);
      pair_elem(baseA + (uint32_t)(c1 + e), av1[e], bv1[e], abar, pc, qa1[e], qb1[e]);
    }
    *(float4*)(q + pr * LDS_STR + c0)       = make_float4(qa0[0], qa0[1], qa0[2], qa0[3]);
    *(float4*)(q + pr * LDS_STR + c1)       = make_float4(qa1[0], qa1[1], qa1[2], qa1[3]);
    *(float4*)(q + (8 + pr) * LDS_STR + c0) = make_float4(qb0[0], qb0[1], qb0[2], qb0[3]);
    *(float4*)(q + (8 + pr) * LDS_STR + c1) = make_float4(qb1[0], qb1[1], qb1[2], qb1[3]);
  }

  // ---- Row sums of the 16 LDS rows ---------------------------------------
  // B-matrix trick: A = ones(16x4) => D[m,n] = sum_k B[k,n]. Lane l supplies
  // B elements of LDS row (l%16); after accumulating over all 64 K-chunks,
  // C-VGPR0 lane l == rowsum(l % 16). All-ones A makes the exact K ordering
  // within the B layout irrelevant.
  const int nrow = lane & 15;
  const int koff = (lane >> 4) << 1;            // 0 or 2
  const float* brow = q + nrow * LDS_STR + koff;
  float sums;
#if __has_builtin(__builtin_amdgcn_wmma_f32_16x16x4_f32)
  {
    v8f acc0 = {};
    v8f acc1 = {};
    v2f ones; ones[0] = 1.0f; ones[1] = 1.0f;
#pragma unroll 8
    for (int c = 0; c < K_CAT; c += 8) {
      v2f bm0 = *(const v2f*)(brow + c);
      v2f bm1 = *(const v2f*)(brow + c + 4);
      acc0 = __builtin_amdgcn_wmma_f32_16x16x4_f32(false, ones, false, bm0,
                                                   (short)0, acc0, false, false);
      acc1 = __builtin_amdgcn_wmma_f32_16x16x4_f32(false, ones, false, bm1,
                                                   (short)0, acc1, false, false);
    }
    sums = acc0[0] + acc1[0];
  }
#else
  {
    float part = 0.0f;
#pragma unroll 8
    for (int c = 0; c < K_CAT; c += 4) {
      v2f bm = *(const v2f*)(brow + c);
      part += bm[0] + bm[1];
    }
    // lanes n and n+16 hold disjoint halves of row n
    sums = part + __shfl_xor(part, 16);
  }
#endif

  // ---- Pass 2: normalize and store ---------------------------------------
#pragma unroll
  for (int lr = 0; lr < 16; ++lr) {
    const float S = __shfl(sums, lr);          // rowsum of local row lr
    const float invS = __builtin_amdgcn_rcpf(S);
    const int gr = (lr < 8) ? (r0 + lr) : (r0 + HALF_ROWS + (lr - 8));
    const uint32_t ob = (uint32_t)gr * K_CAT;
    float4 t0 = *(const float4*)(q + lr * LDS_STR + c0);
    float4 t1 = *(const float4*)(q + lr * LDS_STR + c1);
    t0.x *= invS; t0.y *= invS; t0.z *= invS; t0.w *= invS;
    t1.x *= invS; t1.y *= invS; t1.z *= invS; t1.w *= invS;
    *(float4*)(out + ob + c0) = t0;
    *(float4*)(out + ob + c1) = t1;
  }
}

// ---------------------------------------------------------------------------
extern "C" void kernel_launch(void* const* d_in, const int* in_sizes, int n_in,
                              void* d_out, int out_size, void* d_ws, size_t ws_size,
                              hipStream_t stream) {
  const float* x0   = (const float*)d_in[0];
  const float* beta = (const float*)d_in[1];
  const int*   t    = (const int*)d_in[2];
  float* out = (float*)d_out;
  float* ws  = (float*)d_ws;

  abar_kernel<<<1, 256, 0, stream>>>(beta, t, in_sizes[1], ws);

  const int totalWaves = (ROWS_TOTAL / 2) / PAIRS_PER_WAVE;   // 8192
  const int blocks = totalWaves / WAVES_PER_BLOCK;            // 4096
  const size_t shmem = (size_t)WAVES_PER_BLOCK * 16 * LDS_STR * sizeof(float); // 33280 B
  catdiff_softmax_kernel<<<blocks, WAVES_PER_BLOCK * 32, shmem, stream>>>(x0, ws, out);

  (void)n_in; (void)out_size; (void)ws_size;
}